// GIN_rec_38371237822819
// MI455X (gfx1250) — compile-verified
//
#include <hip/hip_runtime.h>

typedef __attribute__((ext_vector_type(2))) float v2f;
typedef __attribute__((ext_vector_type(8))) float v8f;

#define NNODES 50000
#define NEDGES 600000
#define NGRAPH 64
#define KDIM   128
#define COUT   64
#define AST    132   // LDS A stride (floats): 4-float pad -> conflict-free b64 reads
#define PST    144   // W pair-column stride (float2 units): 2*144%64==32 -> halves disjoint

// ---------------------------------------------------------------- utility
__global__ __launch_bounds__(256) void copy_f32v4(const float* __restrict__ src,
                                                  float* __restrict__ dst, int n4) {
  int i = blockIdx.x * 256 + threadIdx.x;
  if (i < n4) ((float4*)dst)[i] = ((const float4*)src)[i];
}

__global__ __launch_bounds__(256) void zero_f32(float* __restrict__ p, int n) {
  int i = blockIdx.x * 256 + threadIdx.x;
  if (i < n) p[i] = 0.f;
}

// ------------------------------------------------------- edge scatter-add
__global__ __launch_bounds__(256) void edge_scatter(const float* __restrict__ x,
                                                    const int* __restrict__ src,
                                                    const int* __restrict__ dst,
                                                    float* __restrict__ agg, int E) {
  int idx = blockIdx.x * 256 + threadIdx.x;
  int e = idx >> 5;
  if (e >= E) return;
  int lane = idx & 31;
  int ep = e + 2048;
  if (ep < E) __builtin_prefetch(x + (size_t)src[ep] * KDIM + lane * 4, 0, 1);
  int s = src[e], d = dst[e];
  float4 v = *(const float4*)(x + (size_t)s * KDIM + lane * 4);
  float* a = agg + (size_t)d * KDIM + lane * 4;
  atomicAdd(a + 0, v.x); atomicAdd(a + 1, v.y);
  atomicAdd(a + 2, v.z); atomicAdd(a + 3, v.w);
}

// ------------------------------------------------------- global add pool
__global__ __launch_bounds__(256) void pool_scatter(const float* __restrict__ x,
                                                    const int* __restrict__ batch,
                                                    float* __restrict__ pooled, int n) {
  int idx = blockIdx.x * 256 + threadIdx.x;
  int node = idx >> 5;
  if (node >= n) return;
  int lane = idx & 31;
  int b = batch[node];
  float4 v = *(const float4*)(x + (size_t)node * KDIM + lane * 4);
  float* p = pooled + (size_t)b * KDIM + lane * 4;
  atomicAdd(p + 0, v.x); atomicAdd(p + 1, v.y);
  atomicAdd(p + 2, v.z); atomicAdd(p + 3, v.w);
}

// ------------------------------------------------------------ BN stats
__global__ __launch_bounds__(256) void bn_stats(const float* __restrict__ h,
                                                float* __restrict__ sum,
                                                float* __restrict__ sumsq, int n) {
  __shared__ float s0[256], s1[256];
  int tid = threadIdx.x;
  int c = tid & 127;
  int rb = tid >> 7;
  float a = 0.f, a2 = 0.f;
  for (int r = blockIdx.x * 2 + rb; r < n; r += gridDim.x * 2) {
    float v = h[(size_t)r * KDIM + c];
    a += v; a2 += v * v;
  }
  s0[tid] = a; s1[tid] = a2;
  __syncthreads();
  if (tid < 128) {
    atomicAdd(&sum[c],   s0[tid] + s0[tid + 128]);
    atomicAdd(&sumsq[c], s1[tid] + s1[tid + 128]);
  }
}

__global__ void bn_finalize(const float* __restrict__ sum, const float* __restrict__ sumsq,
                            const float* __restrict__ g, const float* __restrict__ b,
                            float* __restrict__ scale, float* __restrict__ shift, int n) {
  int c = threadIdx.x;
  if (c < KDIM) {
    float inv_n = 1.f / (float)n;
    float mu  = sum[c] * inv_n;
    float var = sumsq[c] * inv_n - mu * mu;
    float sc  = g[c] * rsqrtf(var + 1e-5f);
    scale[c] = sc;
    shift[c] = b[c] - mu * sc;
  }
}

// ------------------------------------------------------------ f32 WMMA GEMM
// out[n,m] = act( A'[n,128] @ W[128,m] + bias ),  A' = optional relu(scale*A+shift)
// block = 128 threads (4 wave32); block computes 64 rows x m cols.
template <int MT, bool BN, bool RELU>
__global__ __launch_bounds__(128) void gemm_wmma(
    const float* __restrict__ A, const float* __restrict__ W,
    const float* __restrict__ bias,
    const float* __restrict__ a_scale, const float* __restrict__ a_shift,
    float* __restrict__ out, int n, int m) {
  __shared__ float sA[64 * AST];       // 33 KB, row-major padded
  __shared__ float sW[16 * PST * 2];   // 18 KB, K-pair interleaved: (pair,col,{k0,k1})
  const int tid  = threadIdx.x;
  const int wave = tid >> 5;
  const int lane = tid & 31;
  const int half = lane >> 4;          // K sub-pair select per ISA A/B layout
  const int l16  = lane & 15;
  const int row0 = blockIdx.x * 64;

  // ---- stage A (64 x 128)
  if (!BN) {
    // async DMA path: global -> LDS, 16B per lane, ASYNCcnt tracked.
    // OOB rows clamp to n-1 (same block's rows; results masked at store).
    for (int idx = tid; idx < 64 * 32; idx += 128) {
      int r = idx >> 5, c4 = idx & 31;
      int gr = row0 + r; if (gr > n - 1) gr = n - 1;
      unsigned lds_b = (unsigned)(uintptr_t)(sA + r * AST + c4 * 4);
      unsigned long long ga =
          (unsigned long long)(uintptr_t)(A + (size_t)gr * KDIM + c4 * 4);
      asm volatile("global_load_async_to_lds_b128 %0, %1, off"
                   :: "v"(lds_b), "v"(ga) : "memory");
    }
    asm volatile("s_wait_asynccnt 0x0" ::: "memory");
  } else {
    // manual path with fused BN + ReLU transform
    for (int idx = tid; idx < 64 * 32; idx += 128) {
      int r = idx >> 5, c4 = idx & 31;
      int gr = row0 + r;
      float4 v = make_float4(0.f, 0.f, 0.f, 0.f);
      if (gr < n) {
        v = *(const float4*)(A + (size_t)gr * KDIM + c4 * 4);
        float4 sc = ((const float4*)a_scale)[c4];
        float4 sh = ((const float4*)a_shift)[c4];
        v.x = fmaxf(0.f, v.x * sc.x + sh.x);
        v.y = fmaxf(0.f, v.y * sc.y + sh.y);
        v.z = fmaxf(0.f, v.z * sc.z + sh.z);
        v.w = fmaxf(0.f, v.w * sc.w + sh.w);
      }
      *(float4*)(sA + r * AST + c4 * 4) = v;
    }
  }

  v8f acc[MT] = {};

  for (int kc = 0; kc < KDIM; kc += 32) {
    __syncthreads();
    // ---- stage W chunk (32 x m) pair-interleaved, zero-pad cols >= m
    for (int idx = tid; idx < 32 * 32; idx += 128) {
      int r = idx >> 5, c4 = idx & 31;
      float4 v = make_float4(0.f, 0.f, 0.f, 0.f);
      if (c4 * 4 < m) v = *(const float4*)(W + (size_t)(kc + r) * m + c4 * 4);
      float* dp = sW + ((size_t)(r >> 1) * PST + c4 * 4) * 2 + (r & 1);
      dp[0] = v.x; dp[2] = v.y; dp[4] = v.z; dp[6] = v.w;
    }
    __syncthreads();

    const float* aRow = sA + (size_t)(wave * 16 + l16) * AST + kc + 2 * half;
#pragma unroll
    for (int k = 0; k < 32; k += 4) {
      float2 av = *(const float2*)(aRow + k);   // A frag: rows=lane, K pair by half
      v2f a; a.x = av.x; a.y = av.y;
      const float2* wp = (const float2*)sW + (size_t)((k >> 1) + half) * PST + l16;
#pragma unroll
      for (int t = 0; t < MT; ++t) {
        float2 bv = wp[t * 16];                 // {W[k+2h][col], W[k+2h+1][col]} one b64
        v2f b; b.x = bv.x; b.y = bv.y;
        acc[t] = __builtin_amdgcn_wmma_f32_16x16x4_f32(
            false, a, false, b, (short)0, acc[t], false, false);
      }
    }
  }

  // ---- store: C/D layout -> row = wave*16 + 8*half + vgpr, col = t*16 + l16
#pragma unroll
  for (int t = 0; t < MT; ++t) {
    int col = t * 16 + l16;
    if (col < m) {
      float bv = bias[col];
#pragma unroll
      for (int i = 0; i < 8; ++i) {
        int gr = row0 + wave * 16 + 8 * half + i;
        if (gr < n) {
          float v = acc[t][i] + bv;
          if (RELU) v = fmaxf(v, 0.f);
          out[(size_t)gr * m + col] = v;
        }
      }
    }
  }
}

// ---------------------------------------------------------------- driver
extern "C" void kernel_launch(void* const* d_in, const int* in_sizes, int n_in,
                              void* d_out, int out_size, void* d_ws, size_t ws_size,
                              hipStream_t stream) {
  const float* x      = (const float*)d_in[0];
  const int*   ei     = (const int*)d_in[1];
  const int*   src    = ei;
  const int*   dst    = ei + NEDGES;
  const int*   batch  = (const int*)d_in[2];
  const float* convW1 = (const float*)d_in[3];
  const float* convb1 = (const float*)d_in[4];
  const float* bn_g   = (const float*)d_in[5];
  const float* bn_b   = (const float*)d_in[6];
  const float* convW2 = (const float*)d_in[7];
  const float* convb2 = (const float*)d_in[8];
  const float* recW1  = (const float*)d_in[9];
  const float* recb1  = (const float*)d_in[10];
  const float* recW2  = (const float*)d_in[11];
  const float* recb2  = (const float*)d_in[12];
  const float* recW3  = (const float*)d_in[13];
  const float* recb3  = (const float*)d_in[14];
  const float* mlpW1  = (const float*)d_in[15];
  const float* mlpb1  = (const float*)d_in[16];
  const float* mlpW2  = (const float*)d_in[17];
  const float* mlpb2  = (const float*)d_in[18];

  float* out  = (float*)d_out;             // [64,64]
  float* xrec = out + NGRAPH * COUT;       // [N,4]

  float* ws    = (float*)d_ws;
  float* xcur  = ws;                                   // [N,128]
  float* agg   = ws + (size_t)NNODES * KDIM;           // [N,128] (also h / rec tmp)
  float* stats = ws + 2 * (size_t)NNODES * KDIM;
  float* sum    = stats;
  float* sumsq  = stats + 128;
  float* scale  = stats + 256;
  float* shift  = stats + 384;
  float* pooled = stats + 512;                         // [64,128]
  float* ptmp   = pooled + NGRAPH * KDIM;              // [64,128]

  const int n4         = NNODES * KDIM / 4;
  const int copyBlocks = (n4 + 255) / 256;
  const int scatBlocks = (NEDGES * 32 + 255) / 256;
  const int poolBlocks = (NNODES * 32 + 255) / 256;
  const int gemmBlocks = (NNODES + 63) / 64;

  const float* xin = x;
  for (int l = 0; l < 3; ++l) {
    // agg = x + sum_{j->i} x_j
    copy_f32v4<<<copyBlocks, 256, 0, stream>>>(xin, agg, n4);
    edge_scatter<<<scatBlocks, 256, 0, stream>>>(xin, src, dst, agg, NEDGES);
    zero_f32<<<1, 256, 0, stream>>>(sum, 256);  // sum + sumsq contiguous
    // h = agg @ W1 + b1   (in-place: each block reads/writes only its own rows)
    gemm_wmma<8, false, false><<<gemmBlocks, 128, 0, stream>>>(
        agg, convW1 + (size_t)l * KDIM * KDIM, convb1 + l * KDIM,
        nullptr, nullptr, agg, NNODES, KDIM);
    bn_stats<<<512, 256, 0, stream>>>(agg, sum, sumsq, NNODES);
    bn_finalize<<<1, 128, 0, stream>>>(sum, sumsq, bn_g + l * KDIM, bn_b + l * KDIM,
                                       scale, shift, NNODES);
    // x = relu( relu(bn(h)) @ W2 + b2 )  (BN+ReLU fused into A-load)
    gemm_wmma<8, true, true><<<gemmBlocks, 128, 0, stream>>>(
        agg, convW2 + (size_t)l * KDIM * KDIM, convb2 + l * KDIM,
        scale, shift, xcur, NNODES, KDIM);
    xin = xcur;
  }

  // reconstruction head
  gemm_wmma<8, false, true><<<gemmBlocks, 128, 0, stream>>>(
      xcur, recW1, recb1, nullptr, nullptr, agg, NNODES, KDIM);
  gemm_wmma<8, false, true><<<gemmBlocks, 128, 0, stream>>>(
      agg, recW2, recb2, nullptr, nullptr, agg, NNODES, KDIM);
  gemm_wmma<1, false, true><<<gemmBlocks, 128, 0, stream>>>(
      agg, recW3, recb3, nullptr, nullptr, xrec, NNODES, 4);

  // global add pool + final MLP
  zero_f32<<<(NGRAPH * KDIM + 255) / 256, 256, 0, stream>>>(pooled, NGRAPH * KDIM);
  pool_scatter<<<poolBlocks, 256, 0, stream>>>(xcur, batch, pooled, NNODES);
  gemm_wmma<8, false, true><<<1, 128, 0, stream>>>(
      pooled, mlpW1, mlpb1, nullptr, nullptr, ptmp, NGRAPH, KDIM);
  gemm_wmma<4, false, false><<<1, 128, 0, stream>>>(
      ptmp, mlpW2, mlpb2, nullptr, nullptr, out, NGRAPH, COUT);
}